// Coma4D_C_and_S_13125420057146
// MI455X (gfx1250) — compile-verified
//
#include <hip/hip_runtime.h>

// =====================================================================
// CoMA mesh autoencoder for MI455X (gfx1250, wave32).
// Dense feature transforms: V_WMMA_F32_16X16X4_F32, compile-time
// specialized on (F,G) with zero-padded weight slab staged in LDS.
// Sparse graph propagation: global_atomic_add_f32.
// =====================================================================

typedef __attribute__((ext_vector_type(2))) float v2f;
typedef __attribute__((ext_vector_type(8))) float v8f;

#define PI_F 3.14159265358979323846f

static const int NSL[5]  = {16384, 4096, 1024, 256, 64};
#define BB 2
#define TT 16
#define BTOT 32   // B*T

static inline unsigned gsz(long long total, int block) {
  return (unsigned)((total + block - 1) / block);
}

// ---------------- norm precompute ----------------
__global__ void k_deg(const int* __restrict__ ei, float* __restrict__ deg, int E) {
  int e = blockIdx.x * blockDim.x + threadIdx.x;
  if (e < E) atomicAdd(&deg[ei[e]], 1.0f);   // row = ei[0][e]
}
__global__ void k_dis(const float* __restrict__ deg, float* __restrict__ dis, int N) {
  int i = blockIdx.x * blockDim.x + threadIdx.x;
  if (i < N) {
    float d = deg[i];
    dis[i] = (d > 0.f) ? rsqrtf(fmaxf(d, 1.f)) : 0.f;
  }
}
__global__ void k_wnorm(const int* __restrict__ ei, const float* __restrict__ dis,
                        float* __restrict__ wn, int E) {
  int e = blockIdx.x * blockDim.x + threadIdx.x;
  if (e < E) wn[e] = dis[ei[e]] * dis[ei[E + e]];  // dis[row]*dis[col]
}

// ---------------- phase encoding of input ----------------
__global__ void k_phase(const float* __restrict__ x, float* __restrict__ h, int N) {
  long long i = (long long)blockIdx.x * blockDim.x + threadIdx.x;
  long long total = (long long)BTOT * N * 3;
  if (i >= total) return;
  int c  = (int)(i % 3);
  long long bn = i / 3;
  int n  = (int)(bn % N);
  int bt = (int)(bn / N);
  int t  = bt % TT;
  float th = 2.f * PI_F * (float)t / (float)TT;
  float v = x[((long long)bt * N + n) * 3 + c];
  float* hp = h + ((long long)bt * N + n) * 6;
  hp[c]     = v * __cosf(th);
  hp[c + 3] = v * __sinf(th);
}

// ---------------- sparse propagation (scatter-add) ----------------
__global__ void k_prop(const float* __restrict__ x, const int* __restrict__ ei,
                       const float* __restrict__ wn, float* __restrict__ out,
                       int Bc, int N, int Fd, int E, float scale) {
  long long i = (long long)blockIdx.x * blockDim.x + threadIdx.x;
  long long total = (long long)Bc * E * Fd;
  if (i >= total) return;
  int f = (int)(i % Fd);
  long long be = i / Fd;
  int e = (int)(be % E);
  int b = (int)(be / E);
  int src = ei[e];
  int dst = ei[E + e];
  float msg = scale * wn[e] * x[((long long)b * N + src) * Fd + f];
  atomicAdd(&out[((long long)b * N + dst) * Fd + f], msg);
}

// t2 = t2 - t0   (completes Tk = 2*L*T_{k-1} - T_{k-2})
__global__ void k_sub(float* __restrict__ t2, const float* __restrict__ t0, long long total) {
  long long i = (long long)blockIdx.x * blockDim.x + threadIdx.x;
  if (i < total) t2[i] -= t0[i];
}

// ---------------- WMMA fp32 GEMM accumulate: out[16n x 16g tile] += A(16xF) @ W(FxG)
// Compile-time F,G. Weight slab staged zero-padded in LDS (shared by all 8 waves).
// A layout (32b 16x4): M = lane&15; vgpr pair holds K = k0+2*half, k0+2*half+1.
// B layout (32b 4x16): N = lane&15; same K split across lane halves.
// C/D layout: vgpr r -> (M = r + 8*half, N = lane&15).
template <int F, int G>
__global__ __launch_bounds__(256)
void k_cheb_gemm(const float* __restrict__ Tm, const float* __restrict__ W,
                 float* __restrict__ out, int Bc, int N) {
  constexpr int K4 = (F + 3) & ~3;       // K padded to multiple of 4
  constexpr int GT = (G + 15) / 16;      // 16-wide output tiles
  constexpr int GP = GT * 16;
  __shared__ float Ws[K4][GP];
  for (int i = threadIdx.x; i < K4 * GP; i += blockDim.x) {
    int k = i / GP, g = i % GP;
    Ws[k][g] = (k < F && g < G) ? W[k * G + g] : 0.f;
  }
  __syncthreads();

  int nTiles = N >> 4;
  long long wave  = ((long long)blockIdx.x * blockDim.x + threadIdx.x) >> 5;
  long long total = (long long)Bc * nTiles * GT;
  if (wave >= total) return;
  int lane = threadIdx.x & 31;
  int half = lane >> 4;
  int lm   = lane & 15;
  int gt = (int)(wave % GT);
  long long rest = wave / GT;
  int nt = (int)(rest % nTiles);
  int b  = (int)(rest / nTiles);
  int col  = gt * 16 + lm;                 // output feature column of this lane
  int colc = (col < G) ? col : 0;          // clamped (G=3 tail cols never stored)
  const float* Trow = Tm + ((size_t)b * N + (size_t)nt * 16 + lm) * F;
  float* outT = out + ((size_t)b * N + (size_t)nt * 16) * G;

  v8f c;
#pragma unroll
  for (int r = 0; r < 8; ++r) {
    int m = r + 8 * half;
    c[r] = outT[(size_t)m * G + colc];     // cols >= G: value irrelevant (B padded 0, never stored)
  }
#pragma unroll
  for (int j = 0; j < K4 / 4; ++j) {
    int ka = 4 * j + 2 * half;
    // A fragment: aligned float2 (row base 8B-aligned for F in {6,16,32}, ka even)
    v2f a = *(const v2f*)(Trow + ((F % 4 == 0) ? ka : (ka < F ? ka : 0)));
    if (F % 4 != 0) {                      // zero K-tail (F=6); branch-free select
      bool ok = ka < F;
      a.x = ok ? a.x : 0.f;
      a.y = ok ? a.y : 0.f;
    }
    v2f bm;
    bm.x = Ws[ka][col];                    // zero-padded: unconditional LDS reads
    bm.y = Ws[ka + 1][col];
    c = __builtin_amdgcn_wmma_f32_16x16x4_f32(false, a, false, bm, (short)0, c,
                                              false, false);
  }
#pragma unroll
  for (int r = 0; r < 8; ++r) {
    int m = r + 8 * half;
    if (G % 16 == 0 || col < G) outT[(size_t)m * G + colc] = c[r];
  }
}

static void launch_gemm(hipStream_t s, const float* Tm, const float* W, float* out,
                        int Bc, int N, int Fin, int Fout) {
  const int BK = 256;
  int gT = (Fout + 15) / 16;
  long long thr = (long long)Bc * (N >> 4) * gT * 32;
  unsigned g = gsz(thr, BK);
  if      (Fin == 6  && Fout == 16) k_cheb_gemm< 6, 16><<<g, BK, 0, s>>>(Tm, W, out, Bc, N);
  else if (Fin == 16 && Fout == 16) k_cheb_gemm<16, 16><<<g, BK, 0, s>>>(Tm, W, out, Bc, N);
  else if (Fin == 16 && Fout == 32) k_cheb_gemm<16, 32><<<g, BK, 0, s>>>(Tm, W, out, Bc, N);
  else if (Fin == 32 && Fout == 16) k_cheb_gemm<32, 16><<<g, BK, 0, s>>>(Tm, W, out, Bc, N);
  else if (Fin == 16 && Fout == 3)  k_cheb_gemm<16,  3><<<g, BK, 0, s>>>(Tm, W, out, Bc, N);
}

// ---------------- bias + relu ----------------
__global__ void k_bias_relu(float* __restrict__ out, const float* __restrict__ bias,
                            long long total, int G, int relu) {
  long long i = (long long)blockIdx.x * blockDim.x + threadIdx.x;
  if (i >= total) return;
  float v = out[i];
  if (bias) v += bias[(int)(i % G)];
  if (relu) v = fmaxf(v, 0.f);
  out[i] = v;
}

// ---------------- pooling gather (3 nnz/row) ----------------
__global__ void k_pool(const float* __restrict__ x, const int* __restrict__ idx,
                       const float* __restrict__ w, float* __restrict__ out,
                       int Bc, int Nin, int Nout, int Fd) {
  long long i = (long long)blockIdx.x * blockDim.x + threadIdx.x;
  long long total = (long long)Bc * Nout * Fd;
  if (i >= total) return;
  int f = (int)(i % Fd);
  long long bn = i / Fd;
  int n = (int)(bn % Nout);
  int b = (int)(bn / Nout);
  float acc = 0.f;
#pragma unroll
  for (int k = 0; k < 3; ++k) {
    int j = idx[n * 3 + k];
    acc += w[n * 3 + k] * x[((long long)b * Nin + j) * Fd + f];
  }
  out[((long long)b * Nout + n) * Fd + f] = acc;
}

// ---------------- dense linear: out = in @ W^T (+bias) (+relu), W is [Nout,K]
__global__ void k_linear(const float* __restrict__ in, const float* __restrict__ W,
                         const float* __restrict__ bias, float* __restrict__ out,
                         int M, int N, int K, int relu) {
  int i = blockIdx.x * blockDim.x + threadIdx.x;
  if (i >= M * N) return;
  int n = i % N, m = i / N;
  const float* a = in + (size_t)m * K;
  const float* w = W + (size_t)n * K;
  float acc = bias ? bias[n] : 0.f;
  for (int k = 0; k < K; ++k) acc += a[k] * w[k];
  if (relu) acc = fmaxf(acc, 0.f);
  out[i] = acc;
}

// faithful torch-style time-major mean: out[b',z] = mean_j mu[(j%B), (j/B), zoff+z]
__global__ void k_mumean(const float* __restrict__ mu, float* __restrict__ out,
                         int zoff, int zcnt) {
  int i = blockIdx.x * blockDim.x + threadIdx.x;
  if (i >= BB * zcnt) return;
  int z = i % zcnt, bp = i / zcnt;
  float acc = 0.f;
  for (int j = bp * TT; j < (bp + 1) * TT; ++j) {
    int t = j / BB, b = j % BB;
    acc += mu[((size_t)b * TT + t) * 48 + zoff + z];
  }
  out[i] = acc / (float)TT;
}

// zin[bt, 0:32]=mu_c[b]; [32:48]=mu_s*cos(th_t); [48:64]=mu_s*sin(th_t)
__global__ void k_zin(const float* __restrict__ mu_c, const float* __restrict__ mu_s,
                      float* __restrict__ zin) {
  int i = blockIdx.x * blockDim.x + threadIdx.x;
  if (i >= BTOT * 64) return;
  int cch = i % 64, bt = i / 64;
  int b = bt / TT, t = bt % TT;
  float th = 2.f * PI_F * (float)t / (float)TT;
  float v;
  if (cch < 32)      v = mu_c[b * 32 + cch];
  else if (cch < 48) v = mu_s[b * 16 + (cch - 32)] * __cosf(th);
  else               v = mu_s[b * 16 + (cch - 48)] * __sinf(th);
  zin[i] = v;
}

// s_t = s_avg[:,None] + ds_t
__global__ void k_st(const float* __restrict__ s_avg, const float* __restrict__ ds,
                     float* __restrict__ st, int N) {
  long long i = (long long)blockIdx.x * blockDim.x + threadIdx.x;
  long long total = (long long)BB * TT * N * 3;
  if (i >= total) return;
  int c = (int)(i % 3);
  long long r = i / 3;
  int n  = (int)(r % N);
  int bt = (int)(r / N);
  int b  = bt / TT;
  st[i] = s_avg[((long long)b * N + n) * 3 + c] + ds[i];
}

// ---------------- host-side Chebyshev conv driver ----------------
static void run_cheb(hipStream_t s, float* in, float* out, float* t1buf, float* t2buf,
                     const int* ei, const float* wn, const float* W, const float* bias,
                     int Bc, int N, int E, int Fin, int Fout, int relu) {
  const int BK = 256;
  long long outTot  = (long long)Bc * N * Fout;
  long long tTot    = (long long)Bc * N * Fin;
  long long propTot = (long long)Bc * E * Fin;

  hipMemsetAsync(out, 0, outTot * 4, s);
  // k = 0 : T0 = in
  launch_gemm(s, in, W, out, Bc, N, Fin, Fout);
  // k = 1 : T1 = L x
  hipMemsetAsync(t1buf, 0, tTot * 4, s);
  k_prop<<<gsz(propTot, BK), BK, 0, s>>>(in, ei, wn, t1buf, Bc, N, Fin, E, 1.f);
  launch_gemm(s, t1buf, W + (size_t)Fin * Fout, out, Bc, N, Fin, Fout);
  float* t0 = in; float* t1 = t1buf; float* t2 = t2buf;
  for (int k = 2; k < 6; ++k) {
    hipMemsetAsync(t2, 0, tTot * 4, s);
    k_prop<<<gsz(propTot, BK), BK, 0, s>>>(t1, ei, wn, t2, Bc, N, Fin, E, 2.f);
    k_sub<<<gsz(tTot, BK), BK, 0, s>>>(t2, t0, tTot);
    launch_gemm(s, t2, W + (size_t)k * Fin * Fout, out, Bc, N, Fin, Fout);
    float* tmp = t0; t0 = t1; t1 = t2; t2 = tmp;
  }
  if (bias || relu)
    k_bias_relu<<<gsz(outTot, BK), BK, 0, s>>>(out, bias, outTot, Fout, relu);
}

extern "C" void kernel_launch(void* const* d_in, const int* in_sizes, int n_in,
                              void* d_out, int out_size, void* d_ws, size_t ws_size,
                              hipStream_t stream) {
  (void)in_sizes; (void)n_in; (void)out_size; (void)ws_size;
  const int BK = 256;
  auto Fp = [&](int i) { return (const float*)d_in[i]; };
  auto Ip = [&](int i) { return (const int*)d_in[i]; };

  const float* x = Fp(0);
  const float* eW[4] = {Fp(1), Fp(3), Fp(5), Fp(7)};
  const float* eb[4] = {Fp(2), Fp(4), Fp(6), Fp(8)};
  const float* cW[4] = {Fp(9), Fp(11), Fp(13), Fp(15)};
  const float* cb[3] = {Fp(10), Fp(12), Fp(14)};
  const float* sW[4] = {Fp(16), Fp(18), Fp(20), Fp(22)};
  const float* sb[3] = {Fp(17), Fp(19), Fp(21)};
  const float* muW = Fp(23); const float* mub = Fp(24);
  const float* lincW = Fp(25); const float* lincb = Fp(26);
  const float* linsW = Fp(27); const float* linsb = Fp(28);
  const int* ei[5] = {Ip(29), Ip(30), Ip(31), Ip(32), Ip(33)};
  const int* dsI[4] = {Ip(34), Ip(38), Ip(42), Ip(46)};
  const float* dsW[4] = {Fp(35), Fp(39), Fp(43), Fp(47)};
  const int* usI[4] = {Ip(36), Ip(40), Ip(44), Ip(48)};
  const float* usW[4] = {Fp(37), Fp(41), Fp(45), Fp(49)};

  const int lvlN[5] = {16384, 4096, 1024, 256, 16384};  // graph node counts per edge set
  const int Es[5]   = {12 * 16384, 12 * 4096, 12 * 1024, 12 * 256, 12 * 16384};

  // ---- workspace layout (floats) ----
  float* ws = (float*)d_ws;
  float* dis[5]; float* wn[5];
  size_t off = 0;
  for (int l = 0; l < 5; ++l) { dis[l] = ws + off; off += (size_t)lvlN[l]; }
  float* deg = ws + off; off += 16384;
  for (int l = 0; l < 5; ++l) { wn[l] = ws + off; off += (size_t)Es[l]; }
  float* MU  = ws + off; off += 2048;   // [32,48]
  float* ZIN = ws + off; off += 2048;   // [32,64]
  off = (off + 4095) & ~(size_t)4095;
  const size_t BIG = (size_t)BTOT * 16384 * 16;  // 8.4M floats
  float* bufA = ws + off; off += BIG;
  float* bufB = ws + off; off += BIG;
  float* T1   = ws + off; off += BIG;
  float* T2   = ws + off; off += BIG;

  float* outF  = (float*)d_out;
  float* o_muc = outF;                 // [2,32]
  float* o_mus = outF + 64;            // [2,16]
  float* o_sav = outF + 96;            // [2,16384,3]
  float* o_st  = outF + 96 + (size_t)BB * 16384 * 3;  // [2,16,16384,3]

  // ---- 1. normalized edge weights for all 5 graphs ----
  for (int l = 0; l < 5; ++l) {
    hipMemsetAsync(deg, 0, (size_t)lvlN[l] * 4, stream);
    k_deg<<<gsz(Es[l], BK), BK, 0, stream>>>(ei[l], deg, Es[l]);
    k_dis<<<gsz(lvlN[l], BK), BK, 0, stream>>>(deg, dis[l], lvlN[l]);
    k_wnorm<<<gsz(Es[l], BK), BK, 0, stream>>>(ei[l], dis[l], wn[l], Es[l]);
  }

  // ---- 2. phase-encode input: bufA = [32, 16384, 6] ----
  k_phase<<<gsz((long long)BTOT * 16384 * 3, BK), BK, 0, stream>>>(x, bufA, 16384);

  // ---- 3. encoder: 4x (cheb + downsample) ----
  const int encFi[4] = {6, 16, 16, 16};
  const int encFo[4] = {16, 16, 16, 32};
  for (int l = 0; l < 4; ++l) {
    run_cheb(stream, bufA, bufB, T1, T2, ei[l], wn[l], eW[l], eb[l],
             BTOT, NSL[l], Es[l], encFi[l], encFo[l], 1);
    k_pool<<<gsz((long long)BTOT * NSL[l + 1] * encFo[l], BK), BK, 0, stream>>>(
        bufB, dsI[l], dsW[l], bufA, BTOT, NSL[l], NSL[l + 1], encFo[l]);
  }
  // bufA = [32, 64, 32] == [B, T, 2048]

  // ---- 4. latent ----
  k_linear<<<gsz(BTOT * 48, BK), BK, 0, stream>>>(bufA, muW, mub, MU, BTOT, 48, 2048, 0);
  k_mumean<<<gsz(BB * 32, BK), BK, 0, stream>>>(MU, o_muc, 0, 32);
  k_mumean<<<gsz(BB * 16, BK), BK, 0, stream>>>(MU, o_mus, 32, 16);

  const int decFi[3] = {32, 16, 16};
  const int decFo[3] = {16, 16, 16};

  // ---- 5. decoder_c (batch 2) -> s_avg ----
  k_linear<<<gsz(BB * 2048, BK), BK, 0, stream>>>(o_muc, lincW, lincb, bufA,
                                                  BB, 2048, 32, 0);
  for (int i = 0; i < 3; ++i) {
    int lvl = 3 - i;
    k_pool<<<gsz((long long)BB * NSL[lvl] * decFi[i], BK), BK, 0, stream>>>(
        bufA, usI[lvl], usW[lvl], bufB, BB, NSL[lvl + 1], NSL[lvl], decFi[i]);
    run_cheb(stream, bufB, bufA, T1, T2, ei[lvl], wn[lvl], cW[i], cb[i],
             BB, NSL[lvl], Es[lvl], decFi[i], decFo[i], 1);
  }
  k_pool<<<gsz((long long)BB * NSL[0] * 16, BK), BK, 0, stream>>>(
      bufA, usI[0], usW[0], bufB, BB, NSL[1], NSL[0], 16);
  run_cheb(stream, bufB, o_sav, T1, T2, ei[4], wn[4], cW[3], nullptr,
           BB, NSL[0], Es[4], 16, 3, 0);

  // ---- 6. decoder_s (batch 32) -> ds_t ----
  k_zin<<<gsz(BTOT * 64, BK), BK, 0, stream>>>(o_muc, o_mus, ZIN);
  k_linear<<<gsz(BTOT * 2048, BK), BK, 0, stream>>>(ZIN, linsW, linsb, bufA,
                                                    BTOT, 2048, 64, 1);
  for (int i = 0; i < 3; ++i) {
    int lvl = 3 - i;
    k_pool<<<gsz((long long)BTOT * NSL[lvl] * decFi[i], BK), BK, 0, stream>>>(
        bufA, usI[lvl], usW[lvl], bufB, BTOT, NSL[lvl + 1], NSL[lvl], decFi[i]);
    run_cheb(stream, bufB, bufA, T1, T2, ei[lvl], wn[lvl], sW[i], sb[i],
             BTOT, NSL[lvl], Es[lvl], decFi[i], decFo[i], 1);
  }
  k_pool<<<gsz((long long)BTOT * NSL[0] * 16, BK), BK, 0, stream>>>(
      bufA, usI[0], usW[0], bufB, BTOT, NSL[1], NSL[0], 16);
  run_cheb(stream, bufB, bufA, T1, T2, ei[4], wn[4], sW[3], nullptr,
           BTOT, NSL[0], Es[4], 16, 3, 0);  // ds_t in bufA [32,16384,3]

  // ---- 7. s_t = s_avg + ds_t ----
  k_st<<<gsz((long long)BTOT * 16384 * 3, BK), BK, 0, stream>>>(o_sav, bufA, o_st, 16384);
}